// HypergraphNodeAttentionBlock_28286654612012
// MI455X (gfx1250) — compile-verified
//
#include <hip/hip_runtime.h>
#include <hip/hip_bf16.h>
#include <math.h>

typedef __attribute__((ext_vector_type(2))) float v2f;
typedef __attribute__((ext_vector_type(8))) float v8f;

#define NN 200000
#define GG 16
#define EE 400000
#define HH 400000
#define MM 800000
#define DXX 296      // 128 + 32 + 8 + 64 + 64
#define XP 305       // LDS pitch for x tile (coprime with 64 banks)
#define H1P 265      // LDS pitch for h1 tile
#define LN_EPS 1e-3f

// ---------------------------------------------------------------------------
// prep: fused projection weights (W @ Wc[1]), fused biases, cumsum(n_node)
// ---------------------------------------------------------------------------
__global__ __launch_bounds__(256) void prep_kernel(
    const float* __restrict__ Wq, const float* __restrict__ bq,
    const float* __restrict__ Wk, const float* __restrict__ bk,
    const float* __restrict__ Wc, const float* __restrict__ bc,
    const int* __restrict__ n_node,
    float* __restrict__ wfq, float* __restrict__ bfq,
    float* __restrict__ wfk, float* __restrict__ bfk, int* __restrict__ csum)
{
    const int tid = threadIdx.x;
    const float* Wc1 = Wc + 64;  // Wc[1] : [8][8]
    for (int p = tid; p < 2048; p += 256) {
        const int loc = p & 1023, i = loc >> 3, j = loc & 7;
        const float* W = (p < 1024) ? Wq : Wk;
        float s = 0.f;
#pragma unroll
        for (int t = 0; t < 8; ++t) s += W[i * 8 + t] * Wc1[t * 8 + j];
        ((p < 1024) ? wfq : wfk)[loc] = s;
    }
    if (tid < 8) {
        float sq = 0.f, sk = 0.f;
#pragma unroll
        for (int t = 0; t < 8; ++t) {
            sq += bq[t] * Wc1[t * 8 + tid];
            sk += bk[t] * Wc1[t * 8 + tid];
        }
        bfq[tid] = sq + bc[tid];
        bfk[tid] = sk + bc[tid];
    }
    if (tid == 0) {
        int run = 0;
        for (int g = 0; g < GG; ++g) { run += n_node[g]; csum[g] = run; }
    }
}

// ---------------------------------------------------------------------------
// proj8: dst[r][0..7] = src[r][0..127] @ wf[128][8] + bf   (one wave per row)
// ---------------------------------------------------------------------------
__global__ __launch_bounds__(256) void proj8_kernel(
    const float* __restrict__ src, const float* __restrict__ wf,
    const float* __restrict__ bf, float* __restrict__ dst)
{
    __shared__ float wfs[1024];
    __shared__ float bfs[8];
    const int tid = threadIdx.x;
    for (int p = tid; p < 1024; p += 256) wfs[p] = wf[p];
    if (tid < 8) bfs[tid] = bf[tid];
    __syncthreads();

    const int warp = tid >> 5, lane = tid & 31;
    const size_t row = (size_t)blockIdx.x * 8 + warp;
    const float4 v = *(const float4*)(src + row * 128 + lane * 4);
    const float xv[4] = {v.x, v.y, v.z, v.w};
    float acc[8];
#pragma unroll
    for (int j = 0; j < 8; ++j) acc[j] = 0.f;
    const float* wr = &wfs[lane * 32];
#pragma unroll
    for (int c = 0; c < 4; ++c)
#pragma unroll
        for (int j = 0; j < 8; ++j) acc[j] += xv[c] * wr[c * 8 + j];
#pragma unroll
    for (int off = 16; off > 0; off >>= 1)
#pragma unroll
        for (int j = 0; j < 8; ++j) acc[j] += __shfl_xor(acc[j], off, 32);
#pragma unroll
    for (int j = 0; j < 8; ++j)
        if (lane == j) dst[row * 8 + j] = acc[j] + bfs[j];
}

// ---------------------------------------------------------------------------
// attn: one wave per node; lane k owns incident edge k (K_EDGE == 32)
// softmax over K axis per head, then weighted sum of keys (= values)
// ---------------------------------------------------------------------------
__global__ __launch_bounds__(256) void attn_kernel(
    const int* __restrict__ edge_ind, const float* __restrict__ q_mh,
    const float* __restrict__ k_mh, float* __restrict__ edge_agg)
{
    const int tid = threadIdx.x, warp = tid >> 5, lane = tid & 31;
    const size_t node = (size_t)blockIdx.x * 8 + warp;
    const int eidx = edge_ind[node * 32 + lane];
    const float4 e0 = *(const float4*)(k_mh + (size_t)eidx * 8);
    const float4 e1 = *(const float4*)(k_mh + (size_t)eidx * 8 + 4);
    const float ek[8] = {e0.x, e0.y, e0.z, e0.w, e1.x, e1.y, e1.z, e1.w};
    const float4 q0 = *(const float4*)(q_mh + node * 8);
    const float4 q1 = *(const float4*)(q_mh + node * 8 + 4);
    const float qv[8] = {q0.x, q0.y, q0.z, q0.w, q1.x, q1.y, q1.z, q1.w};
#pragma unroll
    for (int j = 0; j < 8; ++j) {
        const float s = qv[j] * ek[j];
        float m = s;
#pragma unroll
        for (int off = 16; off > 0; off >>= 1) m = fmaxf(m, __shfl_xor(m, off, 32));
        const float e = __expf(s - m);
        float d = e;
#pragma unroll
        for (int off = 16; off > 0; off >>= 1) d += __shfl_xor(d, off, 32);
        float a = (e / d) * ek[j];
#pragma unroll
        for (int off = 16; off > 0; off >>= 1) a += __shfl_xor(a, off, 32);
        if (lane == j) edge_agg[node * 8 + j] = a;
    }
}

// ---------------------------------------------------------------------------
// scatter-add kernels (f32 atomics into L2-resident [N,64] accumulators)
// ---------------------------------------------------------------------------
__global__ __launch_bounds__(256) void hscat_kernel(
    const float* __restrict__ hfeat, const int* __restrict__ hind,
    float* __restrict__ h_agg)
{
    const size_t t = (size_t)blockIdx.x * 256 + threadIdx.x;
    const size_t h = t >> 4;
    const int c = (int)(t & 15);
    const float4 v = *(const float4*)(hfeat + h * 64 + c * 4);
#pragma unroll
    for (int j = 0; j < 3; ++j) {
        const int nd = hind[h * 3 + j];
        float* p = h_agg + (size_t)nd * 64 + c * 4;
        atomicAdd(p + 0, v.x); atomicAdd(p + 1, v.y);
        atomicAdd(p + 2, v.z); atomicAdd(p + 3, v.w);
    }
}

__global__ __launch_bounds__(256) void mscat_kernel(
    const float* __restrict__ mfeat, const int* __restrict__ mind,
    float* __restrict__ m_agg)
{
    const size_t t = (size_t)blockIdx.x * 256 + threadIdx.x;
    const size_t m = t >> 4;
    const int c = (int)(t & 15);
    const float4 v = *(const float4*)(mfeat + m * 64 + c * 4);
    const int nd = mind[m];
    float* p = m_agg + (size_t)nd * 64 + c * 4;
    atomicAdd(p + 0, v.x); atomicAdd(p + 1, v.y);
    atomicAdd(p + 2, v.z); atomicAdd(p + 3, v.w);
}

// ---------------------------------------------------------------------------
// mlp: per block (64 thr / 2 waves) process 32 nodes.
//  x[16,296] per wave in LDS -> WMMA f32 16x16x4 (two independent chains)
//  -> relu -> h1[16,256] in LDS -> WMMA -> relu -> h2[16,128] in LDS
//  -> warp LayerNorm -> out
// ---------------------------------------------------------------------------
__global__ __launch_bounds__(64) void mlp_kernel(
    const float* __restrict__ nodes, const float* __restrict__ globals_,
    const int* __restrict__ csum, const float* __restrict__ edge_agg,
    const float* __restrict__ h_agg, const float* __restrict__ m_agg,
    const float* __restrict__ W1, const float* __restrict__ b1,
    const float* __restrict__ W2, const float* __restrict__ b2,
    const float* __restrict__ ln_g, const float* __restrict__ ln_b,
    float* __restrict__ out)
{
    __shared__ float xs[2][16][XP];     // x tiles (later reused as h2 storage)
    __shared__ float h1s[2][16][H1P];   // h1 tiles
    __shared__ float wpan[296][16];     // weight panel shared by both waves
    __shared__ int cs[16];

    const int tid = threadIdx.x;
    const int wv = tid >> 5;
    const int lane = tid & 31;
    const int half = lane >> 4;
    const int ln16 = lane & 15;
    const int nodeBase = blockIdx.x * 32 + wv * 16;

    if (tid < 16) cs[tid] = csum[tid];
    __syncthreads();

    // ---- stage x = concat(nodes, g_per_node, edge_agg, h_agg, m_agg) ----
    for (int r = 0; r < 16; ++r) {
        const int node = nodeBase + r;
        int gid = 0;
#pragma unroll
        for (int g = 0; g < 16; ++g) gid += (node >= cs[g]) ? 1 : 0;
        for (int c = lane; c < DXX; c += 32) {
            float v;
            if (c < 128)      v = nodes[(size_t)node * 128 + c];
            else if (c < 160) v = globals_[gid * 32 + (c - 128)];
            else if (c < 168) v = edge_agg[(size_t)node * 8 + (c - 160)];
            else if (c < 232) v = h_agg[(size_t)node * 64 + (c - 168)];
            else              v = m_agg[(size_t)node * 64 + (c - 232)];
            xs[wv][r][c] = v;
        }
    }
    __syncthreads();

    // ---- phase 1: h1 = relu(x @ W1 + b1), 16 col-tiles, K = 296 ----
    for (int ct = 0; ct < 16; ++ct) {
        for (int p = tid; p < 296 * 16; p += 64) {
            const int k = p >> 4, n = p & 15;
            wpan[k][n] = W1[k * 256 + ct * 16 + n];
        }
        __syncthreads();
        const float bval = b1[ct * 16 + ln16];
        v8f acc0, acc1;   // two independent WMMA chains (even / odd K-steps)
#pragma unroll
        for (int r = 0; r < 8; ++r) { acc0[r] = bval; acc1[r] = 0.f; }
        for (int i = 0; i < 37; ++i) {   // 74 K-steps, 2 per iteration
            const int kkA = (2 * i) * 4 + half * 2;
            const int kkB = (2 * i + 1) * 4 + half * 2;
            v2f a0, b0, a1, b1v;
            a0.x = xs[wv][ln16][kkA];     a0.y = xs[wv][ln16][kkA + 1];
            b0.x = wpan[kkA][ln16];       b0.y = wpan[kkA + 1][ln16];
            a1.x = xs[wv][ln16][kkB];     a1.y = xs[wv][ln16][kkB + 1];
            b1v.x = wpan[kkB][ln16];      b1v.y = wpan[kkB + 1][ln16];
            acc0 = __builtin_amdgcn_wmma_f32_16x16x4_f32(
                false, a0, false, b0, (short)0, acc0, false, false);
            acc1 = __builtin_amdgcn_wmma_f32_16x16x4_f32(
                false, a1, false, b1v, (short)0, acc1, false, false);
        }
#pragma unroll
        for (int r = 0; r < 8; ++r) {
            float v = acc0[r] + acc1[r];
            h1s[wv][r + half * 8][ct * 16 + ln16] = v > 0.f ? v : 0.f;
        }
        __syncthreads();
    }

    // ---- phase 2: h2 = relu(h1 @ W2 + b2), 8 col-tiles, K = 256 ----
    float* h2w = ((float*)xs) + wv * (16 * 132);  // reuse xs region, pitch 132
    for (int ct = 0; ct < 8; ++ct) {
        for (int p = tid; p < 256 * 16; p += 64) {
            const int k = p >> 4, n = p & 15;
            wpan[k][n] = W2[k * 128 + ct * 16 + n];
        }
        __syncthreads();
        const float bval = b2[ct * 16 + ln16];
        v8f acc0, acc1;
#pragma unroll
        for (int r = 0; r < 8; ++r) { acc0[r] = bval; acc1[r] = 0.f; }
        for (int i = 0; i < 32; ++i) {   // 64 K-steps, 2 per iteration
            const int kkA = (2 * i) * 4 + half * 2;
            const int kkB = (2 * i + 1) * 4 + half * 2;
            v2f a0, b0, a1, b1v;
            a0.x = h1s[wv][ln16][kkA];    a0.y = h1s[wv][ln16][kkA + 1];
            b0.x = wpan[kkA][ln16];       b0.y = wpan[kkA + 1][ln16];
            a1.x = h1s[wv][ln16][kkB];    a1.y = h1s[wv][ln16][kkB + 1];
            b1v.x = wpan[kkB][ln16];      b1v.y = wpan[kkB + 1][ln16];
            acc0 = __builtin_amdgcn_wmma_f32_16x16x4_f32(
                false, a0, false, b0, (short)0, acc0, false, false);
            acc1 = __builtin_amdgcn_wmma_f32_16x16x4_f32(
                false, a1, false, b1v, (short)0, acc1, false, false);
        }
#pragma unroll
        for (int r = 0; r < 8; ++r) {
            float v = acc0[r] + acc1[r];
            h2w[(r + half * 8) * 132 + ct * 16 + ln16] = v > 0.f ? v : 0.f;
        }
        __syncthreads();
    }

    // ---- LayerNorm over 128 features per row (warp-cooperative) ----
    const float4 g4 = ((const float4*)ln_g)[lane];
    const float4 bb4 = ((const float4*)ln_b)[lane];
    for (int m = 0; m < 16; ++m) {
        const float4 v = *(const float4*)(h2w + m * 132 + lane * 4);
        float s = v.x + v.y + v.z + v.w;
        float s2 = v.x * v.x + v.y * v.y + v.z * v.z + v.w * v.w;
#pragma unroll
        for (int off = 16; off > 0; off >>= 1) {
            s += __shfl_xor(s, off, 32);
            s2 += __shfl_xor(s2, off, 32);
        }
        const float mean = s * (1.f / 128.f);
        const float var = s2 * (1.f / 128.f) - mean * mean;
        const float inv = rsqrtf(var + LN_EPS);
        float4 o;
        o.x = (v.x - mean) * inv * g4.x + bb4.x;
        o.y = (v.y - mean) * inv * g4.y + bb4.y;
        o.z = (v.z - mean) * inv * g4.z + bb4.z;
        o.w = (v.w - mean) * inv * g4.w + bb4.w;
        *(float4*)(out + (size_t)(nodeBase + m) * 128 + lane * 4) = o;
    }
}

// ---------------------------------------------------------------------------
extern "C" void kernel_launch(void* const* d_in, const int* in_sizes, int n_in,
                              void* d_out, int out_size, void* d_ws, size_t ws_size,
                              hipStream_t stream)
{
    const float* nodes    = (const float*)d_in[0];
    const float* globals_ = (const float*)d_in[1];
    const int*   n_node   = (const int*)  d_in[2];
    const float* edges    = (const float*)d_in[3];
    const int*   edge_ind = (const int*)  d_in[4];
    const float* hfeat    = (const float*)d_in[5];
    const int*   hind     = (const int*)  d_in[6];
    const float* mfeat    = (const float*)d_in[7];
    const int*   mind     = (const int*)  d_in[8];
    const float* Wq = (const float*)d_in[9];
    const float* bq = (const float*)d_in[10];
    const float* Wk = (const float*)d_in[11];
    const float* bk = (const float*)d_in[12];
    const float* Wc = (const float*)d_in[13];
    const float* bc = (const float*)d_in[14];
    const float* W1 = (const float*)d_in[15];
    const float* b1 = (const float*)d_in[16];
    const float* W2 = (const float*)d_in[17];
    const float* b2 = (const float*)d_in[18];
    const float* ln_g = (const float*)d_in[19];
    const float* ln_b = (const float*)d_in[20];
    float* out = (float*)d_out;

    float* ws = (float*)d_ws;
    int*   csum = (int*)ws;                 // 16 ints
    float* wfq = ws + 16;                   // [128*8]
    float* bfq = ws + 16 + 1024;            // [8]
    float* wfk = ws + 1048;                 // [128*8]
    float* bfk = ws + 2072;                 // [8]
    float* q_mh = ws + 2080;                // [N*8]
    float* k_mh = q_mh + (size_t)NN * 8;    // [E*8]
    float* edge_agg = k_mh + (size_t)EE * 8;    // [N*8]
    float* h_agg = edge_agg + (size_t)NN * 8;   // [N*64]
    float* m_agg = h_agg + (size_t)NN * 64;     // [N*64]

    prep_kernel<<<1, 256, 0, stream>>>(Wq, bq, Wk, bk, Wc, bc, n_node,
                                       wfq, bfq, wfk, bfk, csum);
    hipMemsetAsync(h_agg, 0, (size_t)NN * 128 * sizeof(float), stream);

    proj8_kernel<<<EE / 8, 256, 0, stream>>>(edges, wfk, bfk, k_mh);
    proj8_kernel<<<NN / 8, 256, 0, stream>>>(nodes, wfq, bfq, q_mh);
    attn_kernel<<<NN / 8, 256, 0, stream>>>(edge_ind, q_mh, k_mh, edge_agg);
    hscat_kernel<<<(HH * 16) / 256, 256, 0, stream>>>(hfeat, hind, h_agg);
    mscat_kernel<<<(MM * 16) / 256, 256, 0, stream>>>(mfeat, mind, m_agg);

    mlp_kernel<<<NN / 32, 64, 0, stream>>>(nodes, globals_, csum, edge_agg,
                                           h_agg, m_agg, W1, b1, W2, b2,
                                           ln_g, ln_b, out);
}